// SpGraphAttentionLayer_11330123727204
// MI455X (gfx1250) — compile-verified
//
#include <hip/hip_runtime.h>
#include <stdint.h>

typedef __attribute__((ext_vector_type(2))) float v2f;
typedef __attribute__((ext_vector_type(8))) float v8f;

#define LRELU_SLOPE 0.2f

// Monotone order-preserving float<->uint mapping so we can use the native
// integer atomic max (global_atomic_max_u32) for the fp32 segment max.
__device__ __forceinline__ unsigned enc_f32(float f) {
  unsigned b = __float_as_uint(f);
  return (b & 0x80000000u) ? ~b : (b | 0x80000000u);
}
__device__ __forceinline__ float dec_f32(unsigned e) {
  unsigned b = (e & 0x80000000u) ? (e & 0x7FFFFFFFu) : ~e;
  return __uint_as_float(b);
}

// w_src = W @ a[:d_k], w_dst = W @ a[d_k:]   (W is [in_f, d_k] row-major)
__global__ void wvec_kernel(const float* __restrict__ W, const float* __restrict__ a,
                            float* __restrict__ wsrc, float* __restrict__ wdst,
                            int in_f, int d_k) {
  int i = blockIdx.x * blockDim.x + threadIdx.x;
  if (i >= in_f) return;
  const float* row = W + (size_t)i * d_k;
  float s0 = 0.f, s1 = 0.f;
  for (int j = 0; j < d_k; ++j) {
    float w = row[j];
    s0 = fmaf(w, a[j], s0);
    s1 = fmaf(w, a[d_k + j], s1);
  }
  wsrc[i] = s0;
  wdst[i] = s1;
}

__global__ void init_nodes(unsigned* __restrict__ m_enc, float* __restrict__ denom, int n) {
  int i = blockIdx.x * blockDim.x + threadIdx.x;
  if (i < n) { m_enc[i] = 0u; denom[i] = 0.f; }
}

// Fused dual matvec s_src = x @ w_src, s_dst = x @ w_dst via fp32 WMMA.
// One wave per 16-row tile of x; B is a 4x16 block with cols 0/1 = w vectors.
__global__ void __launch_bounds__(256) matvec_wmma(
    const float* __restrict__ x,
    const float* __restrict__ wsrc, const float* __restrict__ wdst,
    float* __restrict__ ssrc, float* __restrict__ sdst, int n, int tiles) {
  __shared__ __align__(16) float wcat[768];  // [0:256)=wsrc [256:512)=wdst [512:768)=0
  int t = threadIdx.x;
  wcat[t]       = wsrc[t];
  wcat[256 + t] = wdst[t];
  wcat[512 + t] = 0.f;
  __syncthreads();

  int wave = t >> 5;
  int tile = blockIdx.x * 8 + wave;
  if (tile >= tiles) return;          // wave-uniform: EXEC stays all-1s for WMMA

  int lane  = t & 31;
  int col   = lane & 15;              // A row within tile; also D column N
  int khalf = lane >> 4;              // K pair select: lanes 0-15 -> K 0,1 ; 16-31 -> K 2,3
  int row   = tile * 16 + col;
  int lrow  = row < n ? row : n - 1;  // clamp tail loads (stores are guarded)
  const float* xr = x + (size_t)lrow * 256;
  int wsel = col < 2 ? col : 2;       // col 0 -> wsrc, col 1 -> wdst, else zeros
  const v2f* wv = (const v2f*)wcat;

  v8f acc = {};
#pragma unroll 8
  for (int k = 0; k < 256; k += 4) {
    v2f av = *(const v2f*)(xr + k + 2 * khalf);
    v2f bv = wv[(wsel * 256 + k + 2 * khalf) >> 1];
    // D = A(16x4 f32) * B(4x16 f32) + C, fp32 accumulate
    acc = __builtin_amdgcn_wmma_f32_16x16x4_f32(false, av, false, bv,
                                                (short)0, acc, false, false);
  }

  // D layout: vgpr v -> M = v + 8*khalf ; column N = col
  int mbase = tile * 16 + khalf * 8;
  if (col == 0) {
#pragma unroll
    for (int v = 0; v < 8; ++v) { int r = mbase + v; if (r < n) ssrc[r] = acc[v]; }
  } else if (col == 1) {
#pragma unroll
    for (int v = 0; v < 8; ++v) { int r = mbase + v; if (r < n) sdst[r] = acc[v]; }
  }
}

// Pass A: score = leaky_relu(s_src[src] + s_dst[dst]); segment max by src.
__global__ void edge_pass_a(const int* __restrict__ src, const int* __restrict__ dst,
                            const float* __restrict__ ssrc, const float* __restrict__ sdst,
                            float* __restrict__ score, unsigned* __restrict__ m_enc, int E) {
  int e = blockIdx.x * blockDim.x + threadIdx.x;
  if (e >= E) return;
  int s = src[e], d = dst[e];
  float sc = ssrc[s] + sdst[d];
  sc = sc >= 0.f ? sc : LRELU_SLOPE * sc;
  score[e] = sc;
  atomicMax(&m_enc[s], enc_f32(sc));   // native global_atomic_max_u32
}

// Pass B: ex = exp(score - m[src]); segment sum by src (native f32 atomic add).
__global__ void edge_pass_b(const int* __restrict__ src,
                            const float* __restrict__ score_in, float* __restrict__ ex_out,
                            const unsigned* __restrict__ m_enc, float* __restrict__ denom, int E) {
  int e = blockIdx.x * blockDim.x + threadIdx.x;
  if (e >= E) return;
  int s = src[e];
  float ex = __expf(score_in[e] - dec_f32(m_enc[s]));
  ex_out[e] = ex;
  float* addr = denom + s;
  asm volatile("global_atomic_add_f32 %0, %1, off"
               :: "v"((uint64_t)(uintptr_t)addr), "v"(ex)
               : "memory");
}

// Pass C: att = ex / (denom[src] + 1e-16)
__global__ void edge_pass_c(const int* __restrict__ src, const float* __restrict__ ex,
                            const float* __restrict__ denom, float* __restrict__ out, int E) {
  int e = blockIdx.x * blockDim.x + threadIdx.x;
  if (e >= E) return;
  out[e] = ex[e] / (denom[src[e]] + 1e-16f);
}

extern "C" void kernel_launch(void* const* d_in, const int* in_sizes, int n_in,
                              void* d_out, int out_size, void* d_ws, size_t ws_size,
                              hipStream_t stream) {
  const float* x    = (const float*)d_in[0];
  const int*   edge = (const int*)d_in[1];
  const float* W    = (const float*)d_in[2];
  const float* a    = (const float*)d_in[3];
  float* out = (float*)d_out;

  const int IN_F = 256;
  int Nn  = in_sizes[0] / IN_F;       // 100000
  int E   = in_sizes[1] / 2;          // 3200000
  int d_k = in_sizes[3] / 2;          // 128

  // workspace layout (floats/uints)
  float*    wsrc  = (float*)d_ws;
  float*    wdst  = wsrc + IN_F;
  float*    ssrc  = wdst + IN_F;
  float*    sdst  = ssrc + Nn;
  unsigned* m_enc = (unsigned*)(sdst + Nn);
  float*    denom = (float*)(m_enc + Nn);
  float*    score = denom + Nn;       // E floats; reused as ex buffer in pass B

  const int* src = edge;
  const int* dst = edge + E;

  wvec_kernel<<<(IN_F + 255) / 256, 256, 0, stream>>>(W, a, wsrc, wdst, IN_F, d_k);
  init_nodes<<<(Nn + 255) / 256, 256, 0, stream>>>(m_enc, denom, Nn);

  int tiles = (Nn + 15) / 16;
  matvec_wmma<<<(tiles + 7) / 8, 256, 0, stream>>>(x, wsrc, wdst, ssrc, sdst, Nn, tiles);

  int eb = (E + 255) / 256;
  edge_pass_a<<<eb, 256, 0, stream>>>(src, dst, ssrc, sdst, score, m_enc, E);
  edge_pass_b<<<eb, 256, 0, stream>>>(src, score, score, m_enc, denom, E);
  edge_pass_c<<<eb, 256, 0, stream>>>(src, score, denom, out, E);
}